// GCN_20323785244731
// MI455X (gfx1250) — compile-verified
//
#include <hip/hip_runtime.h>
#include <hip/hip_bf16.h>
#include <math.h>

// Problem constants (from reference)
#define BB 8
#define LL 200
#define DD 256
#define HH 8
#define HS 32
#define HB (HH * BB)          // 64
#define KK 20
#define MROWS (BB * LL)       // 1600
#define BLD (BB * LL * DD)    // 409600

#define TPAD 36               // padded tile row (floats): 144B, 16B-aligned

typedef __attribute__((ext_vector_type(2))) float v2f;
typedef __attribute__((ext_vector_type(8))) float v8f;
typedef __attribute__((address_space(3))) float lds_f;

__device__ __forceinline__ float wave_sum32(float v) {
    #pragma unroll
    for (int off = 16; off > 0; off >>= 1) v += __shfl_xor(v, off, 32);
    return v;
}

// Async global->LDS copy of 16B per lane (GLOBAL_LOAD_ASYNC_TO_LDS_B128),
// tracked with ASYNCcnt.
__device__ __forceinline__ void async_copy16(const float* g, float* l) {
    lds_f* lp = (lds_f*)l;   // addrspacecast generic->local: 32-bit LDS address
    asm volatile("global_load_async_to_lds_b128 %0, %1, off"
                 :: "v"(lp), "v"(g) : "memory");
}

__device__ __forceinline__ void wait_async(int n) {
    if (n == 8)      asm volatile("s_wait_asynccnt 0x8" ::: "memory");
    else if (n == 2) asm volatile("s_wait_asynccnt 0x2" ::: "memory");
    else             asm volatile("s_wait_asynccnt 0x0" ::: "memory");
}

// ---------------------------------------------------------------------------
// Kernel 1: out_heads = to_heads(X @ W^T + bias), WMMA f32 16x16x4.
// X: [1600,256], W: [256,256] (row n = output feature), outH: [HB, L, HS]
// grid: (16 tilesN, 100 tilesM), block: 32 (one wave per 16x16 tile)
// ---------------------------------------------------------------------------
__global__ __launch_bounds__(32)
void gemm_heads_wmma(const float* __restrict__ X,
                     const float* __restrict__ W,
                     const float* __restrict__ bias,
                     float* __restrict__ outH) {
    const int tileN = blockIdx.x;
    const int tileM = blockIdx.y;
    const int lane  = threadIdx.x;
    const int half  = lane >> 4;
    const int lr    = lane & 15;

    const int arow = tileM * 16 + lr;
    const int bcol = tileN * 16 + lr;
    const float* ap = X + (size_t)arow * DD;
    const float* wp = W + (size_t)bcol * DD;

    v8f acc = {};
#if __has_builtin(__builtin_amdgcn_wmma_f32_16x16x4_f32)
    for (int kk = 0; kk < DD; kk += 4) {
        v2f a, b;
        a.x = ap[kk + half * 2 + 0];
        a.y = ap[kk + half * 2 + 1];
        b.x = wp[kk + half * 2 + 0];
        b.y = wp[kk + half * 2 + 1];
        acc = __builtin_amdgcn_wmma_f32_16x16x4_f32(
            false, a, false, b, (short)0, acc, false, false);
    }
#else
    for (int r = 0; r < 8; ++r) {
        const int m = tileM * 16 + half * 8 + r;
        const float* xr = X + (size_t)m * DD;
        float s = 0.f;
        for (int kk = 0; kk < DD; ++kk) s = fmaf(xr[kk], wp[kk], s);
        acc[r] = s;
    }
#endif

    const int n  = tileN * 16 + lr;       // C col (N = lr for both halves)
    const int h  = n >> 5;
    const int hs = n & 31;
    const float bv = bias[n];
    #pragma unroll
    for (int r = 0; r < 8; ++r) {
        const int m  = tileM * 16 + half * 8 + r;   // C row
        const int bi = m / LL;
        const int l  = m - bi * LL;
        outH[((size_t)(h * BB + bi) * LL + l) * HS + hs] = acc[r] + bv;
    }
}

// ---------------------------------------------------------------------------
// Kernel 2: raw_graph[hb,q,k] = mask? 0 : dot(a, b+ti) / (|a|*|b+ti| + 1e-6)
// One wave per (hb,q). Transposed scheme: lane = k, loop over hs.
// ti tiles (32 k x 32 hs) staged into LDS via async copy, double-buffered.
// No cross-lane reductions in the hot loop.
// ---------------------------------------------------------------------------
__global__ __launch_bounds__(128)
void scores_kernel(const float* __restrict__ aH, const float* __restrict__ bH,
                   const float* __restrict__ TM, const unsigned char* __restrict__ amask,
                   float* __restrict__ raw) {
    __shared__ __align__(16) float tile[4][2][32 * TPAD];   // 4 waves x 2 bufs x 4.6KB
    const int wid  = threadIdx.x >> 5;
    const int lane = threadIdx.x & 31;
    int rid = blockIdx.x * 4 + wid;
    rid = __builtin_amdgcn_readfirstlane(rid);
    if (rid >= HB * LL) return;
    const int hb = rid / LL, q = rid - hb * LL;
    const int h  = hb >> 3, bi = hb & 7;            // hb = h*B + b, B=8

    const float* arow  = aH + (size_t)rid * HS;     // wave-uniform
    const float* tmrow = TM + ((size_t)(bi * LL + q)) * LL * DD + h * HS;
    float* rrow        = raw + (size_t)rid * LL;
    const unsigned char* mrow = amask + (size_t)q * LL;

    // a[q,:] is uniform across the wave -> scalar loads / SGPR-resident
    float as_[HS];
    float an2 = 0.0f;
    #pragma unroll
    for (int s = 0; s < HS; ++s) { as_[s] = arow[s]; an2 = fmaf(as_[s], as_[s], an2); }
    const float an = sqrtf(an2);

    const int r4 = lane >> 3;           // row-in-group-of-4
    const int c4 = lane & 7;            // 16B chunk within 128B row
    // issue async copies for a chunk of `ng4*4` rows starting at k0 into buf
    auto prefetch = [&](int k0, int buf, int ng4) {
        float* tb = &tile[wid][buf][0];
        #pragma unroll
        for (int j = 0; j < 8; ++j) {
            if (j < ng4) {
                const float* g = tmrow + (size_t)(k0 + 4 * j + r4) * DD + c4 * 4;
                float* l = tb + (4 * j + r4) * TPAD + c4 * 4;
                async_copy16(g, l);
            }
        }
    };

    prefetch(0, 0, 8);
    int buf = 0;
    for (int c = 0; c < 7; ++c) {           // 6 full chunks of 32 k + 1 chunk of 8 k
        int issued = 0;
        if (c < 5)       { prefetch((c + 1) * 32, buf ^ 1, 8); issued = 8; }
        else if (c == 5) { prefetch(192,          buf ^ 1, 2); issued = 2; }
        wait_async(issued);                 // chunk c complete (in-order)

        const int kl = c * 32 + lane;
        const int kb = kl < LL ? kl : (LL - 1);
        const float4* bp = (const float4*)(bH + ((size_t)hb * LL + kb) * HS);
        float4 b4[8];
        #pragma unroll
        for (int j = 0; j < 8; ++j) b4[j] = bp[j];

        const float* trow = &tile[wid][buf][lane * TPAD];
        float s1 = 0.0f, s2 = 0.0f;
        #pragma unroll
        for (int j = 0; j < 8; ++j) {
            const float4 t4 = *(const float4*)(trow + 4 * j);
            float c0 = b4[j].x + t4.x; s1 = fmaf(as_[4*j+0], c0, s1); s2 = fmaf(c0, c0, s2);
            float c1 = b4[j].y + t4.y; s1 = fmaf(as_[4*j+1], c1, s1); s2 = fmaf(c1, c1, s2);
            float c2 = b4[j].z + t4.z; s1 = fmaf(as_[4*j+2], c2, s1); s2 = fmaf(c2, c2, s2);
            float c3 = b4[j].w + t4.w; s1 = fmaf(as_[4*j+3], c3, s1); s2 = fmaf(c3, c3, s2);
        }
        if (kl < LL) {
            rrow[kl] = mrow[kl] ? 0.0f : s1 / (an * sqrtf(s2) + 1e-6f);
        }
        buf ^= 1;
    }
}

// ---------------------------------------------------------------------------
// Kernel 3a: zero the symmetric top-K mask
// ---------------------------------------------------------------------------
__global__ void zero_kernel(float* __restrict__ p, int n) {
    const int i = blockIdx.x * blockDim.x + threadIdx.x;
    if (i < n) p[i] = 0.0f;
}

// ---------------------------------------------------------------------------
// Kernel 3b: top-K=20 per row, set mask[hb,q,idx]=1 and mask[hb,idx,q]=1.
// Register-resident row per wave (7 values/lane), iterative argmax.
// ---------------------------------------------------------------------------
__global__ __launch_bounds__(256)
void topk_mask_kernel(const float* __restrict__ raw, float* __restrict__ maskM) {
    const int wid  = threadIdx.x >> 5;
    const int lane = threadIdx.x & 31;
    const int rid  = blockIdx.x * 8 + wid;
    if (rid >= HB * LL) return;
    const int hb = rid / LL, q = rid - hb * LL;

    const float* rrow = raw + (size_t)rid * LL;
    float v[7];
    #pragma unroll
    for (int j = 0; j < 7; ++j) {
        const int k = lane + 32 * j;
        v[j] = (k < LL) ? rrow[k] : -INFINITY;
    }
    float* mbase = maskM + (size_t)hb * LL * LL;

    for (int it = 0; it < KK; ++it) {
        float bv = -INFINITY;
        int   bk = 0x7fffffff;
        #pragma unroll
        for (int j = 0; j < 7; ++j) {
            const int k = lane + 32 * j;
            if (v[j] > bv) { bv = v[j]; bk = k; }
        }
        #pragma unroll
        for (int off = 16; off > 0; off >>= 1) {
            const float ov = __shfl_xor(bv, off, 32);
            const int   ok = __shfl_xor(bk, off, 32);
            if (ov > bv || (ov == bv && ok < bk)) { bv = ov; bk = ok; }
        }
        if (lane == 0) {
            mbase[(size_t)q * LL + bk] = 1.0f;   // row scatter
            mbase[(size_t)bk * LL + q] = 1.0f;   // symmetric scatter (all write 1.0)
        }
        const int d = bk - lane;
        if (d >= 0 && (d & 31) == 0) v[d >> 5] = -INFINITY;
    }
}

// ---------------------------------------------------------------------------
// Kernel 4: sparse gathers via ballot over nonzero graph entries.
// outputs[hb,q,:] = sum_k s * sH[hb,k,:], time_out[hb,q,:] = sum_k s * ti[hb,q,k,:]
// ---------------------------------------------------------------------------
__global__ __launch_bounds__(256)
void gather_kernel(const float* __restrict__ raw, const float* __restrict__ maskM,
                   const float* __restrict__ sH, const float* __restrict__ TM,
                   float* __restrict__ out_pre, float* __restrict__ tout) {
    const int wid  = threadIdx.x >> 5;
    const int lane = threadIdx.x & 31;
    const int rid  = blockIdx.x * 8 + wid;
    if (rid >= HB * LL) return;
    const int hb = rid / LL, q = rid - hb * LL;
    const int h  = hb >> 3, bi = hb & 7;

    const float* rrow = raw   + (size_t)rid * LL;
    const float* mrow = maskM + (size_t)rid * LL;
    const float* srow = sH    + (size_t)hb * LL * HS;
    const float* trow = TM + ((size_t)(bi * LL + q)) * LL * DD + h * HS;

    float ao = 0.0f, at = 0.0f;
    for (int c = 0; c < 7; ++c) {
        const int kl = c * 32 + lane;
        const float sv = (kl < LL) ? rrow[kl] * mrow[kl] : 0.0f;
        unsigned nz = (unsigned)__ballot(sv != 0.0f);   // wave32: low 32 bits
        while (nz) {
            const int j = __builtin_ctz(nz);
            nz &= nz - 1;
            const float s = __shfl(sv, j, 32);          // uniform across wave
            const int k = c * 32 + j;
            ao = fmaf(s, srow[(size_t)k * HS + lane], ao);
            at = fmaf(s, trow[(size_t)k * DD + lane], at);
        }
    }
    const int od = (bi * LL + q) * DD + h * HS + lane;  // from_heads layout
    out_pre[od] = ao;
    tout[od]    = at;
}

// ---------------------------------------------------------------------------
// Kernel 5: LayerNorm over D=256 per row (biased variance, eps=1e-8)
// ---------------------------------------------------------------------------
__global__ __launch_bounds__(256)
void layernorm_kernel(const float* __restrict__ x,
                      const float* __restrict__ gamma, const float* __restrict__ beta,
                      float* __restrict__ y) {
    __shared__ float red[16];
    const int row = blockIdx.x;
    const int tid = threadIdx.x;
    const float v = x[(size_t)row * DD + tid];

    float s  = wave_sum32(v);
    float s2 = wave_sum32(v * v);
    const int w = tid >> 5;
    if ((tid & 31) == 0) { red[w] = s; red[8 + w] = s2; }
    __syncthreads();
    float ts = 0.0f, ts2 = 0.0f;
    #pragma unroll
    for (int i = 0; i < 8; ++i) { ts += red[i]; ts2 += red[8 + i]; }
    const float mu  = ts * (1.0f / DD);
    const float var = ts2 * (1.0f / DD) - mu * mu;
    const float inv = rsqrtf(var + 1e-8f);
    y[(size_t)row * DD + tid] = (v - mu) * inv * gamma[tid] + beta[tid];
}

// ---------------------------------------------------------------------------
extern "C" void kernel_launch(void* const* d_in, const int* in_sizes, int n_in,
                              void* d_out, int out_size, void* d_ws, size_t ws_size,
                              hipStream_t stream) {
    const float*         seqs    = (const float*)d_in[0];
    const unsigned char* amask   = (const unsigned char*)d_in[1];   // bool [L,L]
    const float*         TM      = (const float*)d_in[2];           // [B,L,L,D]
    const float*         conv1_w = (const float*)d_in[3];
    const float*         conv1_b = (const float*)d_in[4];
    const float*         conv2_w = (const float*)d_in[5];
    const float*         conv2_b = (const float*)d_in[6];
    const float*         W_w     = (const float*)d_in[7];
    const float*         W_b     = (const float*)d_in[8];
    const float*         ln_g    = (const float*)d_in[9];
    const float*         ln_b    = (const float*)d_in[10];

    float* out  = (float*)d_out;          // [0, BLD): ln, [BLD, 2*BLD): time_outputs
    float* ws   = (float*)d_ws;

    float* aH      = ws;                               // HB*L*HS = 409600
    float* bH      = aH + (size_t)HB * LL * HS;        // 409600
    float* sH      = bH + (size_t)HB * LL * HS;        // 409600
    float* raw     = sH + (size_t)HB * LL * HS;        // HB*L*L = 2,560,000
    float* maskM   = raw + (size_t)HB * LL * LL;       // 2,560,000
    float* out_pre = maskM + (size_t)HB * LL * LL;     // 409600

    // 1) Three WMMA GEMMs into head layout
    dim3 ggrid(DD / 16, MROWS / 16);
    gemm_heads_wmma<<<ggrid, 32, 0, stream>>>(seqs, conv1_w, conv1_b, aH);
    gemm_heads_wmma<<<ggrid, 32, 0, stream>>>(seqs, conv2_w, conv2_b, bH);
    gemm_heads_wmma<<<ggrid, 32, 0, stream>>>(seqs, W_w,     W_b,     sH);

    // 2) Scores: bandwidth pass over time_matrices w/ async LDS staging
    const int nrows = HB * LL;                 // 12800 rows, one wave each
    scores_kernel<<<nrows / 4, 128, 0, stream>>>(aH, bH, TM, amask, raw);

    // 3) Top-K symmetric mask
    const int mcount = HB * LL * LL;
    zero_kernel<<<(mcount + 255) / 256, 256, 0, stream>>>(maskM, mcount);
    topk_mask_kernel<<<nrows / 8, 256, 0, stream>>>(raw, maskM);

    // 4) Sparse gathers (outputs + time_outputs)
    gather_kernel<<<nrows / 8, 256, 0, stream>>>(raw, maskM, sH, TM,
                                                 out_pre, out + BLD);

    // 5) LayerNorm
    layernorm_kernel<<<MROWS, 256, 0, stream>>>(out_pre, ln_g, ln_b, out);
}